// CGCN_438086664234
// MI455X (gfx1250) — compile-verified
//
#include <hip/hip_runtime.h>

// LightGCN propagation for MI455X (gfx1250, wave32).
// acc = x0 + A x0 + A^2 x0 + A^3 x0 accumulated in d_out, final *0.25.
// Scatter-add replaced by one-time CSR build (by src) + per-node register gather:
// avoids 384M serialized f32 L2 atomics (the true bottleneck; streaming roofline
// ~140us at 23.3 TB/s, x stays resident in the 192MB L2).
// d_ws layout (~172 MB): Y1, Y2 (76.8MB each), offs[nodes+8], cnt[nodes],
// partials[2048], csr[int2 * nedges].

#define DIM 64

typedef float v2f __attribute__((ext_vector_type(2)));
typedef float v8f __attribute__((ext_vector_type(8)));

// ---------------- init: out = concat(a, b), float4-vectorized ----------------
__global__ void cgcn_copy2(const float4* __restrict__ a, long na4,
                           const float4* __restrict__ b, long nb4,
                           float4* __restrict__ out) {
  long i = (long)blockIdx.x * blockDim.x + threadIdx.x;
  if (i < na4)            out[i] = a[i];
  else if (i < na4 + nb4) out[i] = b[i - na4];
}

// ---------------- zero a buffer, float4-vectorized ----------------
__global__ void cgcn_zero(float4* __restrict__ p, long n4) {
  long i = (long)blockIdx.x * blockDim.x + threadIdx.x;
  if (i < n4) p[i] = make_float4(0.f, 0.f, 0.f, 0.f);
}

// ---------------- CSR build: degree count ----------------
__global__ void cgcn_count(const int* __restrict__ src, int* __restrict__ cnt,
                           int nedges) {
  int e = blockIdx.x * blockDim.x + threadIdx.x;
  if (e < nedges) atomicAdd(&cnt[src[e]], 1);   // global_atomic_add_u32
}

// ---------------- CSR build: block-level exclusive scan ----------------
__global__ void cgcn_scan_block(const int* __restrict__ cnt, int* __restrict__ offs,
                                int* __restrict__ partials, int n) {
  __shared__ int sh[256];
  int t = threadIdx.x;
  int i = blockIdx.x * 256 + t;
  int v = (i < n) ? cnt[i] : 0;
  sh[t] = v;
  __syncthreads();
#pragma unroll
  for (int off = 1; off < 256; off <<= 1) {
    int x = (t >= off) ? sh[t - off] : 0;
    __syncthreads();
    sh[t] += x;
    __syncthreads();
  }
  if (i < n) offs[i] = sh[t] - v;               // exclusive within block
  if (t == 255) partials[blockIdx.x] = sh[255]; // block total
}

// ---------------- CSR build: scan the block totals (single workgroup) ----------------
__global__ void cgcn_scan_partials(int* __restrict__ partials, int nB) {
  __shared__ int sh[256];
  __shared__ int carry;
  int t = threadIdx.x;
  if (t == 0) carry = 0;
  __syncthreads();
  for (int base = 0; base < nB; base += 256) {
    int i = base + t;
    int v = (i < nB) ? partials[i] : 0;
    sh[t] = v;
    __syncthreads();
#pragma unroll
    for (int off = 1; off < 256; off <<= 1) {
      int x = (t >= off) ? sh[t - off] : 0;
      __syncthreads();
      sh[t] += x;
      __syncthreads();
    }
    int incl  = sh[t];
    int total = sh[255];
    int c = carry;
    if (i < nB) partials[i] = c + incl - v;     // exclusive across blocks
    __syncthreads();
    if (t == 0) carry = c + total;
    __syncthreads();
  }
}

// ---------------- CSR build: add block offsets; cap entry ----------------
__global__ void cgcn_scan_add(int* __restrict__ offs, const int* __restrict__ partials,
                              int n, int nedges) {
  int i = blockIdx.x * 256 + threadIdx.x;
  if (i < n) offs[i] += partials[blockIdx.x];
  if (i == 0) offs[n] = nedges;
}

// ---------------- CSR build: place edges (dst, val) packed as int2 ----------------
__global__ void cgcn_place(const int* __restrict__ src, const int* __restrict__ dst,
                           const float* __restrict__ vals, const int* __restrict__ offs,
                           int* __restrict__ cur, int2* __restrict__ csr, int nedges) {
  int e = blockIdx.x * blockDim.x + threadIdx.x;
  if (e >= nedges) return;
  int s = src[e];
  int p = offs[s] + atomicAdd(&cur[s], 1);
  csr[p] = make_int2(dst[e], __float_as_int(vals[e]));
}

// ---------------- per-layer: out[n] = sum_{e in CSR[n]} val_e * x[dst_e] ----------------
// One 16-lane half-wave per node, float4 slice per lane, register accumulation,
// single plain 256B store per node. No float atomics, no pre-zeroing.
__global__ void cgcn_gather(const float* __restrict__ xu,   // rows [0, nu)
                            const float* __restrict__ xi,   // rows [nu, nodes)
                            const int*  __restrict__ offs,
                            const int2* __restrict__ csr,
                            float* __restrict__ out, int nodes, int nu) {
  int n  = (int)((blockIdx.x * blockDim.x + threadIdx.x) >> 4); // halfwave = node
  if (n >= nodes) return;
  int le = threadIdx.x & 15;

  int b = offs[n];
  int e = offs[n + 1];
  float4 acc = make_float4(0.f, 0.f, 0.f, 0.f);
  for (int i = b; i < e; ++i) {
    int2 ed = csr[i];
    float v = __int_as_float(ed.y);
    const float* row = (ed.x < nu) ? (xu + (size_t)ed.x * DIM)
                                   : (xi + (size_t)(ed.x - nu) * DIM);
    float4 g = *(const float4*)(row + le * 4);
    acc.x += g.x * v; acc.y += g.y * v; acc.z += g.z * v; acc.w += g.w * v;
  }
  *(float4*)(out + (size_t)n * DIM + le * 4) = acc;
}

// ---------------- WMMA tile op: dst = src*(scaleB*I16) + (useC ? dst : 0) ----------------
// 16x16 tiles, one wave32/tile, 4 chained V_WMMA_F32_16X16X4_F32 (K=16 in K=4 steps).
// A 16x4 f32 layout (ISA 7.12.2): lane L: M=L&15, half h=L>>4: VGPR0=A[M,2h], VGPR1=A[M,2h+1].
// B = scaled identity rows 4j..4j+3; C/D: VGPR r: lanes0-15 M=r, lanes16-31 M=r+8, N=lane&15.
__global__ void cgcn_wmma_affine(const float* __restrict__ src,
                                 float* __restrict__ dst,
                                 float scaleB, int useC, int nTiles) {
  int wave = (int)((blockIdx.x * blockDim.x + threadIdx.x) >> 5);
  if (wave >= nTiles) return;          // wave-uniform: EXEC stays all-1s

  int lane = threadIdx.x & 31;
  int h = lane >> 4;
  int m = lane & 15;

  int colTile = wave & (DIM / 16 - 1);
  long rowBase = (long)(wave >> 2) * 16;
  int c0 = colTile * 16;

  const float* sA = src + rowBase * DIM + c0;
  float*       dA = dst + rowBase * DIM + c0;

  const float* arow = sA + (size_t)m * DIM + 2 * h;
  v2f a0 = *(const v2f*)(arow + 0);
  v2f a1 = *(const v2f*)(arow + 4);
  v2f a2 = *(const v2f*)(arow + 8);
  v2f a3 = *(const v2f*)(arow + 12);

  v2f b0 = { (m == 2 * h + 0)  ? scaleB : 0.f, (m == 2 * h + 1)  ? scaleB : 0.f };
  v2f b1 = { (m == 2 * h + 4)  ? scaleB : 0.f, (m == 2 * h + 5)  ? scaleB : 0.f };
  v2f b2 = { (m == 2 * h + 8)  ? scaleB : 0.f, (m == 2 * h + 9)  ? scaleB : 0.f };
  v2f b3 = { (m == 2 * h + 12) ? scaleB : 0.f, (m == 2 * h + 13) ? scaleB : 0.f };

  v8f c = {};
  if (useC) {
#pragma unroll
    for (int r = 0; r < 8; ++r)
      c[r] = dA[(size_t)(r + 8 * h) * DIM + m];
  }

  c = __builtin_amdgcn_wmma_f32_16x16x4_f32(false, a0, false, b0, (short)0, c, false, false);
  c = __builtin_amdgcn_wmma_f32_16x16x4_f32(false, a1, false, b1, (short)0, c, false, false);
  c = __builtin_amdgcn_wmma_f32_16x16x4_f32(false, a2, false, b2, (short)0, c, false, false);
  c = __builtin_amdgcn_wmma_f32_16x16x4_f32(false, a3, false, b3, (short)0, c, false, false);

#pragma unroll
  for (int r = 0; r < 8; ++r)
    dA[(size_t)(r + 8 * h) * DIM + m] = c[r];
}

extern "C" void kernel_launch(void* const* d_in, const int* in_sizes, int n_in,
                              void* d_out, int out_size, void* d_ws, size_t ws_size,
                              hipStream_t stream) {
  const float* user_emb  = (const float*)d_in[0];
  const float* item_emb  = (const float*)d_in[1];
  const float* edge_vals = (const float*)d_in[2];
  const int*   edge_src  = (const int*)d_in[3];
  const int*   edge_dst  = (const int*)d_in[4];

  const long nu    = in_sizes[0] / DIM;       // 100000
  const long ni    = in_sizes[1] / DIM;       // 200000
  const long nodes = nu + ni;                 // 300000
  const int  nedges = in_sizes[2];            // 2000000

  float* out = (float*)d_out;

  // workspace carve-up (16B-aligned sections; nodes % 4 == 0)
  float* Y1   = (float*)d_ws;
  float* Y2   = Y1 + nodes * DIM;
  int*   offs = (int*)(Y2 + nodes * DIM);     // nodes+1 used, nodes+8 reserved
  int*   cnt  = offs + (nodes + 8);
  int*   partials = cnt + nodes;              // 2048 reserved
  int2*  csr  = (int2*)(partials + 2048);

  const int TB = 256;
  const long tot4   = nodes * DIM / 4;
  const int  cpBlk  = (int)((tot4 + TB - 1) / TB);
  const int  edBlk  = (nedges + TB - 1) / TB;
  const int  nB     = (int)((nodes + 255) / 256);             // scan blocks
  const int  gaBlk  = (int)((nodes * 16 + TB - 1) / TB);
  const int  nTiles = (int)(nodes / 16) * (DIM / 16);         // 75000
  const int  wmBlk  = (int)(((long)nTiles * 32 + TB - 1) / TB);

  // acc = x0
  cgcn_copy2<<<cpBlk, TB, 0, stream>>>((const float4*)user_emb, in_sizes[0] / 4,
                                       (const float4*)item_emb, in_sizes[1] / 4,
                                       (float4*)out);

  // ---- CSR build (once per launch) ----
  cgcn_zero<<<(int)(nodes / 4 + TB - 1) / TB, TB, 0, stream>>>((float4*)cnt, nodes / 4);
  cgcn_count<<<edBlk, TB, 0, stream>>>(edge_src, cnt, nedges);
  cgcn_scan_block<<<nB, 256, 0, stream>>>(cnt, offs, partials, (int)nodes);
  cgcn_scan_partials<<<1, 256, 0, stream>>>(partials, nB);
  cgcn_scan_add<<<nB, 256, 0, stream>>>(offs, partials, (int)nodes, nedges);
  cgcn_zero<<<(int)(nodes / 4 + TB - 1) / TB, TB, 0, stream>>>((float4*)cnt, nodes / 4);
  cgcn_place<<<edBlk, TB, 0, stream>>>(edge_src, edge_dst, edge_vals, offs, cnt, csr, nedges);

  // ---- 3 propagation layers: Y = A @ x (CSR gather), acc += Y (WMMA) ----
  cgcn_gather<<<gaBlk, TB, 0, stream>>>(user_emb, item_emb, offs, csr, Y1,
                                        (int)nodes, (int)nu);
  cgcn_wmma_affine<<<wmBlk, TB, 0, stream>>>(Y1, out, 1.0f, 1, nTiles);

  cgcn_gather<<<gaBlk, TB, 0, stream>>>(Y1, Y1 + nu * DIM, offs, csr, Y2,
                                        (int)nodes, (int)nu);
  cgcn_wmma_affine<<<wmBlk, TB, 0, stream>>>(Y2, out, 1.0f, 1, nTiles);

  cgcn_gather<<<gaBlk, TB, 0, stream>>>(Y2, Y2 + nu * DIM, offs, csr, Y1,
                                        (int)nodes, (int)nu);
  cgcn_wmma_affine<<<wmBlk, TB, 0, stream>>>(Y1, out, 1.0f, 1, nTiles);

  // final: out = acc * 0.25 (WMMA scaled identity, C = 0)
  cgcn_wmma_affine<<<wmBlk, TB, 0, stream>>>(out, out, 0.25f, 0, nTiles);
}